// GNEpropGIN_79285096284500
// MI455X (gfx1250) — compile-verified
//
#include <hip/hip_runtime.h>

// ---------------------------------------------------------------------------
// GNEprop-GIN forward for MI455X (gfx1250, wave32).
// GEMMs: V_WMMA_F32_16X16X4_F32, one wave computes a 16x64 output strip
// (4 accumulator tiles): each A fragment feeds 4 WMMAs. K-loop is branchless
// with strength-reduced (pointer-increment) addressing; ragged-K tail uses
// clamped addresses + v_cndmask zeroing of the A fragment only.
// Message passing scatter uses global_atomic_add_f32 (unsafeAtomicAdd).
// ---------------------------------------------------------------------------

typedef float v2f __attribute__((ext_vector_type(2)));
typedef float v8f __attribute__((ext_vector_type(8)));

#define WAVES_PER_BLOCK 4

// Problem constants (match reference)
#define CN   50000
#define CE   800000
#define CB   2000
#define CIN  133
#define CED  14
#define CH   64
#define CL   5
#define CMF  200
#define CFFN 512
#define CJK  (CH * (CL + 1))      // 384
#define CHMOL (CJK + CMF)         // 584

// ---------------------------------------------------------------------------
// WMMA GEMM:  C[M x Nc] = A[M x K] @ W[K x Nc] + bias
// One wave32 computes a 16x64 strip = 4 16x16 tiles. Requires Nc % 64 == 0,
// M % 16 == 0 (true for every call here).
// f32 WMMA layouts (ISA 7.12.2):
//   A 16x4 : a[v] = A[lane%16, 2*(lane/16)+v]
//   B 4x16 : b[v] = B[2*(lane/16)+v, lane%16]
//   D 16x16: d[v] = D[v + 8*(lane/16), lane%16]
// ---------------------------------------------------------------------------
__global__ void wmma_gemm_bias(const float* __restrict__ A, int lda, int aoff,
                               const float* __restrict__ W,
                               const float* __restrict__ bias,
                               float* __restrict__ C, int ldc, int coff,
                               int M, int K, int Nc) {
  const int groupsN = Nc >> 6;                       // 64-wide column groups
  const int totalGroups = (M >> 4) * groupsN;
  const int grp = blockIdx.x * WAVES_PER_BLOCK + (threadIdx.x >> 5);
  if (grp >= totalGroups) return;                    // wave-uniform
  const int tm = (grp / groupsN) << 4;
  const int tn = (grp % groupsN) << 6;
  const int lane = threadIdx.x & 31;
  const int half = lane >> 4;                        // 0: lanes 0-15, 1: 16-31
  const int l16  = lane & 15;
  const int kh   = half << 1;                        // per-lane K sub-offset

  // Loop-invariant per-lane base pointers; only induction adds in the loop.
  const float* __restrict__ arow = A + (size_t)(tm + l16) * lda + aoff + kh;
  const float* __restrict__ r0   = W + (size_t)kh * Nc + tn + l16;  // row kh
  const float* __restrict__ r1   = r0 + Nc;                         // row kh+1
  const size_t wstep = (size_t)Nc * 4;               // rows advance by 4 per step

  v8f acc0 = {}, acc1 = {}, acc2 = {}, acc3 = {};
  const int Kmain = K & ~3;

  int k = 0;
  for (; k < Kmain; k += 4) {
    v2f a;
    a.x = arow[k];
    a.y = arow[k + 1];
    v2f b0, b1, b2, b3;
    b0.x = r0[0];  b0.y = r1[0];
    b1.x = r0[16]; b1.y = r1[16];
    b2.x = r0[32]; b2.y = r1[32];
    b3.x = r0[48]; b3.y = r1[48];
    acc0 = __builtin_amdgcn_wmma_f32_16x16x4_f32(false, a, false, b0, (short)0, acc0, false, false);
    acc1 = __builtin_amdgcn_wmma_f32_16x16x4_f32(false, a, false, b1, (short)0, acc1, false, false);
    acc2 = __builtin_amdgcn_wmma_f32_16x16x4_f32(false, a, false, b2, (short)0, acc2, false, false);
    acc3 = __builtin_amdgcn_wmma_f32_16x16x4_f32(false, a, false, b3, (short)0, acc3, false, false);
    r0 += wstep;
    r1 += wstep;
  }

  if (k < K) {                                       // ragged-K tail (K%4 = 1 or 2)
    const int ka  = k + kh;
    const int ka0 = (ka     < K) ? ka     : (K - 1); // clamp: loads always legal
    const int ka1 = (ka + 1 < K) ? ka + 1 : (K - 1);
    const float* __restrict__ abase = A + (size_t)(tm + l16) * lda + aoff;
    v2f a;
    a.x = (ka     < K) ? abase[ka0] : 0.0f;          // v_cndmask, no branch
    a.y = (ka + 1 < K) ? abase[ka1] : 0.0f;
    // a==0 kills the product, so B can use clamped rows without selects.
    const float* __restrict__ t0 = W + (size_t)ka0 * Nc + tn + l16;
    const float* __restrict__ t1 = W + (size_t)ka1 * Nc + tn + l16;
    v2f b0, b1, b2, b3;
    b0.x = t0[0];  b0.y = t1[0];
    b1.x = t0[16]; b1.y = t1[16];
    b2.x = t0[32]; b2.y = t1[32];
    b3.x = t0[48]; b3.y = t1[48];
    acc0 = __builtin_amdgcn_wmma_f32_16x16x4_f32(false, a, false, b0, (short)0, acc0, false, false);
    acc1 = __builtin_amdgcn_wmma_f32_16x16x4_f32(false, a, false, b1, (short)0, acc1, false, false);
    acc2 = __builtin_amdgcn_wmma_f32_16x16x4_f32(false, a, false, b2, (short)0, acc2, false, false);
    acc3 = __builtin_amdgcn_wmma_f32_16x16x4_f32(false, a, false, b3, (short)0, acc3, false, false);
  }

  float* __restrict__ crow = C + coff + tn + l16;
  const float bv0 = bias[tn + l16 +  0];
  const float bv1 = bias[tn + l16 + 16];
  const float bv2 = bias[tn + l16 + 32];
  const float bv3 = bias[tn + l16 + 48];
#pragma unroll
  for (int v = 0; v < 8; ++v) {
    const size_t r = (size_t)(tm + (half << 3) + v) * ldc;
    crow[r +  0] = acc0[v] + bv0;
    crow[r + 16] = acc1[v] + bv1;
    crow[r + 32] = acc2[v] + bv2;
    crow[r + 48] = acc3[v] + bv3;
  }
}

// ---------------------------------------------------------------------------
__global__ void zero_f32(float* __restrict__ p, long n) {
  long i = (long)blockIdx.x * blockDim.x + threadIdx.x;
  const long stride = (long)gridDim.x * blockDim.x;
  for (; i < n; i += stride) p[i] = 0.0f;
}

// msg = relu(h[src] + e); agg[dst] += msg   (one thread per (edge, 4 channels))
__global__ void edge_msg_scatter(const float* __restrict__ hjk, int lda, int hoff,
                                 const float* __restrict__ e,
                                 const int* __restrict__ src,
                                 const int* __restrict__ dst,
                                 float* __restrict__ agg, long total4) {
  const long i = (long)blockIdx.x * blockDim.x + threadIdx.x;
  if (i >= total4) return;
  const long edge = i >> 4;                 // CH/4 = 16 quads per edge
  const int  c4   = (int)(i & 15) << 2;
  const float4 ev = *(const float4*)(e + ((size_t)edge << 6) + c4);
  const float4 hv = *(const float4*)(hjk + (size_t)src[edge] * lda + hoff + c4);
  float* __restrict__ ap = agg + ((size_t)dst[edge] << 6) + c4;
  unsafeAtomicAdd(ap + 0, fmaxf(hv.x + ev.x, 0.0f));
  unsafeAtomicAdd(ap + 1, fmaxf(hv.y + ev.y, 0.0f));
  unsafeAtomicAdd(ap + 2, fmaxf(hv.z + ev.z, 0.0f));
  unsafeAtomicAdd(ap + 3, fmaxf(hv.w + ev.w, 0.0f));
}

// z = (1+eps)*h + agg   (vectorized, CH=64)
__global__ void gin_combine(const float* __restrict__ hjk, int lda, int hoff,
                            const float* __restrict__ agg,
                            const float* __restrict__ eps_p,
                            float* __restrict__ z, long total4) {
  const long i = (long)blockIdx.x * blockDim.x + threadIdx.x;
  if (i >= total4) return;
  const long row = i >> 4;
  const int  c4  = (int)(i & 15) << 2;
  const float s = 1.0f + eps_p[0];
  const float4 hv = *(const float4*)(hjk + (size_t)row * lda + hoff + c4);
  const float4 av = *(const float4*)(agg + ((size_t)row << 6) + c4);
  float4 r;
  r.x = s * hv.x + av.x; r.y = s * hv.y + av.y;
  r.z = s * hv.z + av.z; r.w = s * hv.w + av.w;
  *(float4*)(z + ((size_t)row << 6) + c4) = r;
}

// Coalesced per-column partial sums: blockDim.x == C, thread c owns column c,
// block owns a contiguous row chunk; one float atomic per column per block.
__global__ void bn_partial(const float* __restrict__ t, int M, int C,
                           float* __restrict__ gsum, float* __restrict__ gsq,
                           int rowsPerBlock) {
  const int c  = threadIdx.x;
  const int r0 = blockIdx.x * rowsPerBlock;
  const int r1 = (r0 + rowsPerBlock < M) ? (r0 + rowsPerBlock) : M;
  float sum = 0.0f, sq = 0.0f;
  for (int r = r0; r < r1; ++r) {
    const float v = t[(size_t)r * C + c];
    sum += v; sq += v * v;
  }
  unsafeAtomicAdd(&gsum[c], sum);
  unsafeAtomicAdd(&gsq[c], sq);
}

__global__ void bn_finalize(const float* __restrict__ gsum, const float* __restrict__ gsq,
                            int M, int C,
                            const float* __restrict__ g, const float* __restrict__ be,
                            float* __restrict__ scale, float* __restrict__ shift) {
  const int c = blockIdx.x * blockDim.x + threadIdx.x;
  if (c >= C) return;
  const float mu  = gsum[c] / (float)M;
  const float var = gsq[c] / (float)M - mu * mu;
  const float sc  = g[c] * rsqrtf(var + 1e-5f);
  scale[c] = sc;
  shift[c] = be[c] - mu * sc;
}

// y[row*ldy + yoff + c] = relu(x[row*C + c]*scale[c] + shift[c])  (vectorized)
__global__ void bn_apply_relu(const float* __restrict__ x, int C,
                              const float* __restrict__ scale,
                              const float* __restrict__ shift,
                              float* __restrict__ y, int ldy, int yoff, long total4) {
  const long i = (long)blockIdx.x * blockDim.x + threadIdx.x;
  if (i >= total4) return;
  const int  cq  = C >> 2;
  const long row = i / cq;
  const int  c4  = (int)(i - row * cq) << 2;
  const float4 xv = *(const float4*)(x + (size_t)row * C + c4);
  const float4 sc = *(const float4*)(scale + c4);
  const float4 sh = *(const float4*)(shift + c4);
  float4 r;
  r.x = fmaxf(xv.x * sc.x + sh.x, 0.0f);
  r.y = fmaxf(xv.y * sc.y + sh.y, 0.0f);
  r.z = fmaxf(xv.z * sc.z + sh.z, 0.0f);
  r.w = fmaxf(xv.w * sc.w + sh.w, 0.0f);
  *(float4*)(y + (size_t)row * ldy + yoff + c4) = r;
}

// pooled[batch[row]] += hjk[row]   (one thread per (node, 4 channels))
__global__ void pool_scatter(const float* __restrict__ hjk,
                             const int* __restrict__ batch,
                             float* __restrict__ pooled, long total4) {
  const long i = (long)blockIdx.x * blockDim.x + threadIdx.x;
  if (i >= total4) return;
  const long row = i / (CJK / 4);
  const int  c4  = (int)(i - row * (CJK / 4)) << 2;
  const float4 hv = *(const float4*)(hjk + (size_t)row * CJK + c4);
  float* __restrict__ pp = pooled + (size_t)batch[row] * CJK + c4;
  unsafeAtomicAdd(pp + 0, hv.x);
  unsafeAtomicAdd(pp + 1, hv.y);
  unsafeAtomicAdd(pp + 2, hv.z);
  unsafeAtomicAdd(pp + 3, hv.w);
}

__global__ void count_nodes(const int* __restrict__ batch, float* __restrict__ cnt, int N) {
  const int i = blockIdx.x * blockDim.x + threadIdx.x;
  if (i < N) unsafeAtomicAdd(&cnt[batch[i]], 1.0f);
}

// outc[b, :384] = pooled[b]/max(cnt,1);  outc[b, 384:584] = mol_features[b]
__global__ void pool_finalize(const float* __restrict__ pooled,
                              const float* __restrict__ cnt,
                              const float* __restrict__ molf,
                              float* __restrict__ outc, long total) {
  const long i = (long)blockIdx.x * blockDim.x + threadIdx.x;
  if (i >= total) return;
  const long b = i / CHMOL;
  const int  c = (int)(i - b * CHMOL);
  if (c < CJK) {
    outc[i] = pooled[(size_t)b * CJK + c] / fmaxf(cnt[b], 1.0f);
  } else {
    outc[i] = molf[(size_t)b * CMF + (c - CJK)];
  }
}

// final [B,512] @ [512,1] + b
__global__ void final_linear(const float* __restrict__ r, const float* __restrict__ w,
                             const float* __restrict__ b, float* __restrict__ out, int B, int K) {
  const int i = blockIdx.x * blockDim.x + threadIdx.x;
  if (i >= B) return;
  const float* __restrict__ row = r + (size_t)i * K;
  float acc = 0.0f;
  for (int k = 0; k < K; k += 4) {
    const float4 rv = *(const float4*)(row + k);
    const float4 wv = *(const float4*)(w + k);
    acc += rv.x * wv.x + rv.y * wv.y + rv.z * wv.z + rv.w * wv.w;
  }
  out[i] = acc + b[0];
}

// ---------------------------------------------------------------------------
// Input leaf indices: recursive insertion-order flattening of setup_inputs():
//  0:x 1:edge_attr 2:mol_features
//  3:node_W 4:node_b 5:edge_W 6:edge_b
//  7+9l .. 15+9l : conv l = {eps, W1, b1, g1, be1, W2, b2, g2, be2}
//  52..55: readout0 {W,b,g,be}   56..59: readout1 {W,b,g,be}
//  60:final_W 61:final_b 62:edge_index 63:batch
// ---------------------------------------------------------------------------
extern "C" void kernel_launch(void* const* d_in, const int* in_sizes, int n_in,
                              void* d_out, int out_size, void* d_ws, size_t ws_size,
                              hipStream_t stream) {
  (void)in_sizes; (void)n_in; (void)out_size; (void)ws_size;

  const float* x         = (const float*)d_in[0];
  const float* edge_attr = (const float*)d_in[1];
  const float* molf      = (const float*)d_in[2];
  const float* node_W    = (const float*)d_in[3];
  const float* node_b    = (const float*)d_in[4];
  const float* edge_W    = (const float*)d_in[5];
  const float* edge_b    = (const float*)d_in[6];
  const int*   edge_index= (const int*)d_in[62];
  const int*   batch     = (const int*)d_in[63];
  const int*   src = edge_index;        // edge_index[0, :]
  const int*   dst = edge_index + CE;   // edge_index[1, :]

  // Workspace layout (floats), 256B-aligned chunks
  float* ws = (float*)d_ws;
  size_t off = 0;
  auto alloc = [&](size_t n) { float* p = ws + off; off += (n + 63) & ~(size_t)63; return p; };
  float* e_emb  = alloc((size_t)CE * CH);     // edge embeddings [E,64]
  float* hjk    = alloc((size_t)CN * CJK);    // JK-concat node states [N,384]
  float* agg    = alloc((size_t)CN * CH);
  float* zb     = alloc((size_t)CN * CH);
  float* t1     = alloc((size_t)CN * 2 * CH);
  float* z1     = alloc((size_t)CN * 2 * CH);
  float* t2     = alloc((size_t)CN * CH);
  float* pooled = alloc((size_t)CB * CJK);
  float* cnt    = alloc((size_t)CB);
  float* outc   = alloc((size_t)CB * CHMOL);
  float* r1t    = alloc((size_t)CB * CFFN);
  float* r1     = alloc((size_t)CB * CFFN);
  float* r2t    = alloc((size_t)CB * CFFN);
  float* r2     = alloc((size_t)CB * CFFN);
  float* scale  = alloc(CFFN);
  float* shift  = alloc(CFFN);
  float* gsum   = alloc(CFFN);
  float* gsq    = alloc(CFFN);

  auto gemm = [&](const float* A, int lda, int aoff, const float* W, const float* bias,
                  float* C, int ldc, int coff, int M, int K, int Nc) {
    const int groups = (M >> 4) * (Nc >> 6);
    const int blocks = (groups + WAVES_PER_BLOCK - 1) / WAVES_PER_BLOCK;
    wmma_gemm_bias<<<blocks, 32 * WAVES_PER_BLOCK, 0, stream>>>(
        A, lda, aoff, W, bias, C, ldc, coff, M, K, Nc);
  };
  auto bn = [&](const float* t, int M, int C, const float* g, const float* be,
                float* y, int ldy, int yoff) {
    zero_f32<<<4, 256, 0, stream>>>(gsum, C);
    zero_f32<<<4, 256, 0, stream>>>(gsq, C);
    const int rpb = 256;
    bn_partial<<<(M + rpb - 1) / rpb, C, 0, stream>>>(t, M, C, gsum, gsq, rpb);
    bn_finalize<<<(C + 255) / 256, 256, 0, stream>>>(gsum, gsq, M, C, g, be, scale, shift);
    const long total4 = (long)M * C / 4;
    bn_apply_relu<<<(int)((total4 + 255) / 256), 256, 0, stream>>>(
        t, C, scale, shift, y, ldy, yoff, total4);
  };

  // --- encoders ---
  gemm(x,         CIN, 0, node_W, node_b, hjk,   CJK, 0, CN, CIN, CH);  // h0 -> hjk[:,0:64]
  gemm(edge_attr, CED, 0, edge_W, edge_b, e_emb, CH,  0, CE, CED, CH);  // e  -> [E,64]

  // --- 5 GIN conv layers ---
  for (int l = 0; l < CL; ++l) {
    const int base = 7 + 9 * l;
    const float* eps = (const float*)d_in[base + 0];
    const float* W1  = (const float*)d_in[base + 1];
    const float* b1  = (const float*)d_in[base + 2];
    const float* g1  = (const float*)d_in[base + 3];
    const float* be1 = (const float*)d_in[base + 4];
    const float* W2  = (const float*)d_in[base + 5];
    const float* b2  = (const float*)d_in[base + 6];
    const float* g2  = (const float*)d_in[base + 7];
    const float* be2 = (const float*)d_in[base + 8];

    zero_f32<<<2048, 256, 0, stream>>>(agg, (long)CN * CH);
    const long etotal4 = (long)CE * (CH / 4);
    edge_msg_scatter<<<(int)((etotal4 + 255) / 256), 256, 0, stream>>>(
        hjk, CJK, CH * l, e_emb, src, dst, agg, etotal4);
    const long ntotal4 = (long)CN * (CH / 4);
    gin_combine<<<(int)((ntotal4 + 255) / 256), 256, 0, stream>>>(
        hjk, CJK, CH * l, agg, eps, zb, ntotal4);

    gemm(zb, CH,     0, W1, b1, t1, 2 * CH, 0, CN, CH,     2 * CH);  // [N,64]->[N,128]
    bn(t1, CN, 2 * CH, g1, be1, z1, 2 * CH, 0);
    gemm(z1, 2 * CH, 0, W2, b2, t2, CH,     0, CN, 2 * CH, CH);      // [N,128]->[N,64]
    bn(t2, CN, CH, g2, be2, hjk, CJK, CH * (l + 1));                 // h_{l+1} -> hjk slot
  }

  // --- global mean pool + concat mol features ---
  zero_f32<<<512, 256, 0, stream>>>(pooled, (long)CB * CJK);
  zero_f32<<<8, 256, 0, stream>>>(cnt, (long)CB);
  const long ptotal4 = (long)CN * (CJK / 4);
  pool_scatter<<<(int)((ptotal4 + 255) / 256), 256, 0, stream>>>(hjk, batch, pooled, ptotal4);
  count_nodes<<<(CN + 255) / 256, 256, 0, stream>>>(batch, cnt, CN);
  const long ototal = (long)CB * CHMOL;
  pool_finalize<<<(int)((ototal + 255) / 256), 256, 0, stream>>>(pooled, cnt, molf, outc, ototal);

  // --- readout FFN ---
  const float* r0W  = (const float*)d_in[52];
  const float* r0b  = (const float*)d_in[53];
  const float* r0g  = (const float*)d_in[54];
  const float* r0be = (const float*)d_in[55];
  const float* r1W  = (const float*)d_in[56];
  const float* r1b  = (const float*)d_in[57];
  const float* r1g  = (const float*)d_in[58];
  const float* r1be = (const float*)d_in[59];
  const float* fW   = (const float*)d_in[60];
  const float* fb   = (const float*)d_in[61];

  gemm(outc, CHMOL, 0, r0W, r0b, r1t, CFFN, 0, CB, CHMOL, CFFN);
  bn(r1t, CB, CFFN, r0g, r0be, r1, CFFN, 0);
  gemm(r1, CFFN, 0, r1W, r1b, r2t, CFFN, 0, CB, CFFN, CFFN);
  bn(r2t, CB, CFFN, r1g, r1be, r2, CFFN, 0);

  final_linear<<<(CB + 255) / 256, 256, 0, stream>>>(r2, fW, fb, (float*)d_out, CB, CFFN);
}